// GAT_link_prediction_74431783240462
// MI455X (gfx1250) — compile-verified
//
#include <hip/hip_runtime.h>
#include <hip/hip_bf16.h>

// ---------------- problem constants (from reference setup_inputs) ------------
#define NN      50000           // nodes
#define NE      800000          // edges (without self loops)
#define NQ      100000          // link queries
#define ETOT    (NE + NN)       // edges incl. self loops = 850000
#define NEG_SLOPE 0.2f
#define EPSV     1e-16f
// key encoding of -inf for monotonic uint atomicMax on floats
#define KEY_NEG_INF 0x007FFFFFu

typedef __attribute__((ext_vector_type(2))) float v2f;
typedef __attribute__((ext_vector_type(8))) float v8f;

static __device__ __forceinline__ unsigned fkey(float f) {
    unsigned u = __float_as_uint(f);
    return (u & 0x80000000u) ? ~u : (u | 0x80000000u);
}
static __device__ __forceinline__ float funkey(unsigned k) {
    return __uint_as_float((k & 0x80000000u) ? (k & 0x7fffffffu) : ~k);
}

// ---------------------------- init kernels -----------------------------------
__global__ void fill_f32(float* __restrict__ p, float v, int n) {
    int t = blockIdx.x * blockDim.x + threadIdx.x;
    if (t < n) p[t] = v;
}
__global__ void fill_u32(unsigned* __restrict__ p, unsigned v, int n) {
    int t = blockIdx.x * blockDim.x + threadIdx.x;
    if (t < n) p[t] = v;
}

// -------------------- fp32 WMMA GEMM: C[M,N] = act(A)[M,K] @ B[K,N] ----------
// One wave per 16x16 output tile, K-loop in steps of 4 using V_WMMA_F32_16X16X4_F32.
// A-operand layout (ISA 7.12.2, 32-bit A 16x4): lanes 0-15 -> K = k0,k0+1 ;
// lanes 16-31 -> K = k0+2,k0+3 ; M = lane&15.  B mirrors (N = lane&15).
// C/D: VGPR v holds M = v + 8*(lane>=16), N = lane&15.
// RELU_BIAS=true: A element is relu(A + biasA[k]) (fuses the layer-1 epilogue
// into the layer-2 GEMM A-load). Compile-time flag -> straight-line K-loop.
template <bool RELU_BIAS>
__global__ void gemm_wmma_f32(const float* __restrict__ A,
                              const float* __restrict__ B,
                              float* __restrict__ C,
                              const float* __restrict__ biasA,
                              int M, int K, int Nout)
{
    int wid  = (blockIdx.x * blockDim.x + threadIdx.x) >> 5;
    int lane = threadIdx.x & 31;
    int tilesN  = Nout >> 4;
    int nTiles  = (M >> 4) * tilesN;
    if (wid >= nTiles) return;            // wave-uniform: EXEC stays all-ones for WMMA
    int mbase = (wid / tilesN) << 4;
    int nbase = (wid % tilesN) << 4;
    int lrow  = lane & 15;
    int khalf = lane >> 4;                // 0 or 1
    int k2    = khalf << 1;               // 0 or 2

    // running pointers: no per-iteration 64-bit multiplies
    const float* aptr = A + (size_t)(mbase + lrow) * K + k2;   // 8B aligned (k2,K even)
    const float* bptr = B + (size_t)k2 * Nout + nbase + lrow;
    const float* cptr = RELU_BIAS ? (biasA + k2) : nullptr;
    const int bstep = Nout << 2;          // 4 rows of B per iteration

    v8f acc = {};
#pragma unroll 4
    for (int k0 = 0; k0 < K; k0 += 4) {
        v2f av = *(const v2f*)aptr;       // K = k0+k2, k0+k2+1 (contiguous b64 load)
        if (RELU_BIAS) {
            v2f bb = *(const v2f*)cptr;
            av.x = fmaxf(av.x + bb.x, 0.0f);
            av.y = fmaxf(av.y + bb.y, 0.0f);
            cptr += 4;
        }
        v2f bv;
        bv.x = bptr[0];                   // row k0+k2
        bv.y = bptr[Nout];                // row k0+k2+1
        acc = __builtin_amdgcn_wmma_f32_16x16x4_f32(
                  false, av, false, bv, (short)0, acc, false, false);
        aptr += 4;
        bptr += bstep;
    }
    int orow = mbase + (khalf << 3);
#pragma unroll
    for (int v = 0; v < 8; ++v)
        C[(size_t)(orow + v) * Nout + nbase + lrow] = acc[v];
}

// ------------- per-node attention coefficients: alpha[n,h] = <h[n,h,:], a[h,:]>
__global__ void alpha_kernel(const float* __restrict__ hmat,
                             const float* __restrict__ a_src,
                             const float* __restrict__ a_dst,
                             float* __restrict__ as, float* __restrict__ ad,
                             int H, int Cdim)
{
    int t = blockIdx.x * blockDim.x + threadIdx.x;
    if (t >= NN * H) return;
    int n = t / H, h = t % H;
    const float* row = hmat + (size_t)n * H * Cdim + (size_t)h * Cdim;
    const float* vs  = a_src + h * Cdim;
    const float* vd  = a_dst + h * Cdim;
    float ss = 0.f, sd = 0.f;
    for (int c = 0; c < Cdim; ++c) {
        float v = row[c];
        ss += v * vs[c];
        sd += v * vd[c];
    }
    as[t] = ss;
    ad[t] = sd;
}

// ---------------- edge logits + leaky relu + segment max (uint-key atomics) ---
__global__ void edge_logits_max(const int* __restrict__ ei,
                                const float* __restrict__ as,
                                const float* __restrict__ ad,
                                float* __restrict__ elog,
                                unsigned* __restrict__ mkey, int H)
{
    int e = blockIdx.x * blockDim.x + threadIdx.x;
    if (e >= ETOT) return;
    int s, d;
    if (e < NE) { s = ei[e]; d = ei[NE + e]; }
    else        { s = d = e - NE; }
    for (int h = 0; h < H; ++h) {
        float v = as[s * H + h] + ad[d * H + h];
        v = (v >= 0.f) ? v : NEG_SLOPE * v;
        elog[(size_t)e * H + h] = v;
        atomicMax(&mkey[d * H + h], fkey(v));
    }
}

// ---------------- exp(e - m[dst]) + segment sum ------------------------------
__global__ void edge_exp_sum(const int* __restrict__ ei,
                             const unsigned* __restrict__ mkey,
                             float* __restrict__ elog,     // in: logit, out: exp
                             float* __restrict__ ssum, int H)
{
    int e = blockIdx.x * blockDim.x + threadIdx.x;
    if (e >= ETOT) return;
    int d = (e < NE) ? ei[NE + e] : (e - NE);
    for (int h = 0; h < H; ++h) {
        float m = funkey(mkey[d * H + h]);
        if (!__builtin_isfinite(m)) m = 0.f;   // matches jnp.where(isfinite(m), m, 0)
        float ex = expf(elog[(size_t)e * H + h] - m);
        elog[(size_t)e * H + h] = ex;
        atomicAdd(&ssum[d * H + h], ex);
    }
}

// -------- weighted message scatter-add: agg[dst] += h[src] * w (wave/edge) ----
__global__ void edge_aggregate(const int* __restrict__ ei,
                               const float* __restrict__ hmat,
                               const float* __restrict__ ex,
                               const float* __restrict__ ssum,
                               float* __restrict__ agg,
                               int H, int Cdim, int OUT)
{
    int gid  = blockIdx.x * blockDim.x + threadIdx.x;
    int e    = gid >> 5;
    int lane = gid & 31;
    if (e >= ETOT) return;
    int s, d;
    if (e < NE) { s = ei[e]; d = ei[NE + e]; }
    else        { s = d = e - NE; }
    const float* hrow = hmat + (size_t)s * OUT;
    float*       arow = agg  + (size_t)d * OUT;
    for (int j = lane; j < OUT; j += 32) {
        int h   = j / Cdim;
        float w = ex[(size_t)e * H + h] / (ssum[d * H + h] + EPSV);
        atomicAdd(&arow[j], hrow[j] * w);
    }
}

// ---------------- z2 += b2 ----------------------------------------------------
__global__ void add_bias32(float* __restrict__ z, const float* __restrict__ b) {
    int t = blockIdx.x * blockDim.x + threadIdx.x;
    if (t < NN * 32) z[t] += b[t & 31];
}

// ---------------- decode: out[q] = <z2[i0], z2[i1]> (32 floats) --------------
__global__ void decode_dot(const int* __restrict__ eli,
                           const float* __restrict__ z2,
                           float* __restrict__ out)
{
    int q = blockIdx.x * blockDim.x + threadIdx.x;
    if (q >= NQ) return;
    int i0 = eli[q];
    int i1 = eli[NQ + q];
    const float4* p0 = (const float4*)(z2 + (size_t)i0 * 32);
    const float4* p1 = (const float4*)(z2 + (size_t)i1 * 32);
    float acc = 0.f;
#pragma unroll
    for (int k = 0; k < 8; ++k) {
        float4 a = p0[k], b = p1[k];
        acc += a.x * b.x + a.y * b.y + a.z * b.z + a.w * b.w;
    }
    out[q] = acc;
}

// ----------------------------------------------------------------------------
static inline int cdiv(long long a, long long b) { return (int)((a + b - 1) / b); }

extern "C" void kernel_launch(void* const* d_in, const int* in_sizes, int n_in,
                              void* d_out, int out_size, void* d_ws, size_t ws_size,
                              hipStream_t stream) {
    const float* x      = (const float*)d_in[0];
    const int*   ei     = (const int*)  d_in[1];   // [2, NE] int32 (jax x64 off)
    const int*   eli    = (const int*)  d_in[2];   // [2, NQ]
    const float* W1     = (const float*)d_in[3];   // [128,256]
    const float* a1s    = (const float*)d_in[4];   // [4,64]
    const float* a1d    = (const float*)d_in[5];
    const float* b1     = (const float*)d_in[6];   // [256]
    const float* W2     = (const float*)d_in[7];   // [256,32]
    const float* a2s    = (const float*)d_in[8];   // [2,16]
    const float* a2d    = (const float*)d_in[9];
    const float* b2     = (const float*)d_in[10];  // [32]
    float*       out    = (float*)d_out;

    // ---- workspace layout (floats) ----
    float* ws = (float*)d_ws;
    size_t off = 0;
    float* h1  = ws + off; off += (size_t)NN * 256;   // x @ W1
    float* z1  = ws + off; off += (size_t)NN * 256;   // layer-1 aggregate
    float* h2  = ws + off; off += (size_t)NN * 32;    // relu(z1+b1) @ W2
    float* z2  = ws + off; off += (size_t)NN * 32;    // layer-2 aggregate (+b2)
    float* as1 = ws + off; off += (size_t)NN * 4;
    float* ad1 = ws + off; off += (size_t)NN * 4;
    float* as2 = ws + off; off += (size_t)NN * 2;
    float* ad2 = ws + off; off += (size_t)NN * 2;
    unsigned* mkey = (unsigned*)(ws + off); off += (size_t)NN * 4;  // reused L2 (H=2)
    float* ssum = ws + off; off += (size_t)NN * 4;                  // reused L2
    float* exb  = ws + off; off += (size_t)ETOT * 4;                // reused L2

    const int B = 256;

    // =================== layer 1 (H=4, C=64, OUT=256) ===================
    fill_f32<<<cdiv((size_t)NN * 256, B), B, 0, stream>>>(z1, 0.f, NN * 256);
    fill_u32<<<cdiv(NN * 4, B), B, 0, stream>>>(mkey, KEY_NEG_INF, NN * 4);
    fill_f32<<<cdiv(NN * 4, B), B, 0, stream>>>(ssum, 0.f, NN * 4);

    {   // h1 = x @ W1 : 3125 x 16 tiles, 8 waves/block
        int tiles = (NN / 16) * (256 / 16);
        gemm_wmma_f32<false><<<cdiv((size_t)tiles * 32, B), B, 0, stream>>>(
            x, W1, h1, nullptr, NN, 128, 256);
    }
    alpha_kernel<<<cdiv(NN * 4, B), B, 0, stream>>>(h1, a1s, a1d, as1, ad1, 4, 64);
    edge_logits_max<<<cdiv(ETOT, B), B, 0, stream>>>(ei, as1, ad1, exb, mkey, 4);
    edge_exp_sum<<<cdiv(ETOT, B), B, 0, stream>>>(ei, mkey, exb, ssum, 4);
    edge_aggregate<<<cdiv((size_t)ETOT * 32, B), B, 0, stream>>>(
        ei, h1, exb, ssum, z1, 4, 64, 256);

    // =================== layer 2 (H=2, C=16, OUT=32) ====================
    fill_f32<<<cdiv(NN * 32, B), B, 0, stream>>>(z2, 0.f, NN * 32);
    fill_u32<<<cdiv(NN * 2, B), B, 0, stream>>>(mkey, KEY_NEG_INF, NN * 2);
    fill_f32<<<cdiv(NN * 2, B), B, 0, stream>>>(ssum, 0.f, NN * 2);

    {   // h2 = relu(z1 + b1) @ W2 : 3125 x 2 tiles
        int tiles = (NN / 16) * (32 / 16);
        gemm_wmma_f32<true><<<cdiv((size_t)tiles * 32, B), B, 0, stream>>>(
            z1, W2, h2, b1, NN, 256, 32);
    }
    alpha_kernel<<<cdiv(NN * 2, B), B, 0, stream>>>(h2, a2s, a2d, as2, ad2, 2, 16);
    edge_logits_max<<<cdiv(ETOT, B), B, 0, stream>>>(ei, as2, ad2, exb, mkey, 2);
    edge_exp_sum<<<cdiv(ETOT, B), B, 0, stream>>>(ei, mkey, exb, ssum, 2);
    edge_aggregate<<<cdiv((size_t)ETOT * 32, B), B, 0, stream>>>(
        ei, h2, exb, ssum, z2, 2, 16, 32);
    add_bias32<<<cdiv(NN * 32, B), B, 0, stream>>>(z2, b2);

    // =================== decode =========================================
    decode_dot<<<cdiv(NQ, B), B, 0, stream>>>(eli, z2, out);
}